// Net_81716047774006
// MI455X (gfx1250) — compile-verified
//
#include <hip/hip_runtime.h>

typedef __attribute__((ext_vector_type(2))) float v2f;
typedef __attribute__((ext_vector_type(8))) float v8f;

#define N_ANCH 9

// Stage 1: streaming kernel. Each float4 = two (w,h) boxes.
// Writes iou1 per element and one float partial sum per block.
__global__ __launch_bounds__(256) void box_iou_kernel(
    const float* __restrict__ boxes,    // n_boxes*2 floats, interleaved (w,h)
    const float* __restrict__ anchors,  // 9*2 floats
    float* __restrict__ out_iou1,       // d_out + 1 (4B aligned only)
    float* __restrict__ partials,       // one partial per block
    int n4)                             // number of float4 chunks = n_boxes/2
{
    __shared__ float4 tab[N_ANCH * 2];  // [2k]   = {w_t, h_t, w_t*h_t, 0}
                                        // [2k+1] = {w_t1, h_t1, w_t1*h_t1, 0}
    __shared__ float sdata[256];

    const int t = threadIdx.x;
    if (t < N_ANCH) {
        float ax = anchors[2 * t + 0] * 6.0f;
        float ay = anchors[2 * t + 1] * 6.0f;
        float wt  = expf(ax);
        float ht  = expf(ay);
        float wt1 = rintf(wt);          // round-half-even == jnp.round
        float ht1 = rintf(ht);
        tab[2 * t + 0] = make_float4(wt,  ht,  wt  * ht,  0.0f);
        tab[2 * t + 1] = make_float4(wt1, ht1, wt1 * ht1, 0.0f);
    }
    __syncthreads();

    const float4* __restrict__ in4 = (const float4*)boxes;
    const int stride = gridDim.x * blockDim.x;
    float acc = 0.0f;

    for (int i = blockIdx.x * blockDim.x + t; i < n4; i += stride) {
        float4 v = in4[i];
        // Prefetch next tile this thread will touch (speculative; OOB is dropped).
        __builtin_prefetch(&in4[i + stride], 0, 0);
#pragma unroll
        for (int k = 0; k < 2; ++k) {
            float w = k ? v.z : v.x;
            float h = k ? v.w : v.y;
            float mx    = fmaxf(w, h);
            float ratio = w / h;
            int base = (mx > 190.0f) ? 0 : ((mx > 94.0f) ? 3 : 6);
            int off  = (ratio > 2.0f) ? 2 : ((ratio < 0.5f) ? 1 : 0);
            int key  = base + off;

            float4 t0 = tab[2 * key + 0];
            float4 t1 = tab[2 * key + 1];

            float area   = w * h;
            float inter  = fminf(w, t0.x) * fminf(h, t0.y);
            float invio  = (area + t0.z - inter) / inter;      // 1/iou
            float inter1 = fminf(w, t1.x) * fminf(h, t1.y);
            float iou1   = inter1 / (area + t1.z - inter1);

            out_iou1[2 * i + k] = iou1;
            acc += invio;
        }
    }

    // Deterministic block reduction (LDS tree).
    sdata[t] = acc;
    __syncthreads();
#pragma unroll
    for (int s = 128; s > 0; s >>= 1) {
        if (t < s) sdata[t] += sdata[t + s];
        __syncthreads();
    }
    if (t == 0) partials[blockIdx.x] = sdata[0];
}

// Stage 2: single wave32 (EXEC all ones -> WMMA legal). Reduces the block
// partials. The 32->16 halving step is done with v_wmma_f32_16x16x4_f32:
// A = ones(16x4), B[0][j] = p[j] (lanes 0..15), B[2][j] = p[j+16] (lanes
// 16..31), B rows 1,3 = 0  =>  D[0][j] = p[j] + p[j+16], exact f32 adds.
__global__ __launch_bounds__(32) void final_reduce(
    const float* __restrict__ partials, int n,
    float* __restrict__ out0, float inv_n)
{
    const int lane = threadIdx.x;  // 0..31, one wave
    float acc = 0.0f;
    for (int i = lane; i < n; i += 32) acc += partials[i];

    v2f a; a[0] = 1.0f; a[1] = 1.0f;   // A-matrix: all ones
    v2f b; b[0] = acc;  b[1] = 0.0f;   // B-matrix: partials in row 0/2
    v8f c = {};
    v8f d = __builtin_amdgcn_wmma_f32_16x16x4_f32(
        /*neg_a=*/false, a, /*neg_b=*/false, b,
        /*c_mod=*/(short)0, c, /*reuse_a=*/false, /*reuse_b=*/false);

    // d[0] in lanes 0..15 holds the 16 column sums p[j]+p[j+16].
    float s = d[0];
    float total = 0.0f;
#pragma unroll
    for (int j = 0; j < 16; ++j) total += __shfl(s, j, 32);

    if (lane == 0) out0[0] = total * inv_n;
}

extern "C" void kernel_launch(void* const* d_in, const int* in_sizes, int n_in,
                              void* d_out, int out_size, void* d_ws, size_t ws_size,
                              hipStream_t stream) {
    const float* boxes   = (const float*)d_in[0];
    const float* anchors = (const float*)d_in[1];
    float* out      = (float*)d_out;        // out[0]=scalar, out[1..N]=iou1
    float* partials = (float*)d_ws;

    const int n_boxes = in_sizes[0] / 2;    // 8,388,608
    const int n4      = n_boxes / 2;        // float4 chunks

    const int threads = 256;
    const int blocks  = 4096;               // 16 KB of partials in d_ws

    box_iou_kernel<<<blocks, threads, 0, stream>>>(boxes, anchors,
                                                   out + 1, partials, n4);
    final_reduce<<<1, 32, 0, stream>>>(partials, blocks, out,
                                       1.0f / (float)n_boxes);
}